// SimpleHGN_3221225472215
// MI455X (gfx1250) — compile-verified
//
#include <hip/hip_runtime.h>
#include <math.h>

// ---------------------------------------------------------------------------
// SimpleHGN forward on gfx1250 (MI455X). f32 WMMA for all dense GEMMs,
// double-buffered async global->LDS tile staging (ASYNCcnt-pipelined),
// atomic-based edge softmax + scatter-add for the graph aggregation.
// ---------------------------------------------------------------------------

typedef float v2f __attribute__((ext_vector_type(2)));
typedef float v8f __attribute__((ext_vector_type(8)));

#define GEMM_WAVES 8          // 256 threads / block, one 16x16 C tile per wave
#define ALPHA_C 0.05f
#define SLOPE_C 0.2f

// issue one 16x16 A-tile + 16x16 B-tile as 4 x (32 lanes x b128) async loads
__device__ __forceinline__ void async_tile_load(
    unsigned lA0, unsigned lA1, unsigned lB0, unsigned lB1,
    const float* aP0, const float* aP1, const float* bP0, const float* bP1)
{
    asm volatile("global_load_async_to_lds_b128 %0, %1, off"
                 :: "v"(lA0), "v"(aP0) : "memory");
    asm volatile("global_load_async_to_lds_b128 %0, %1, off"
                 :: "v"(lA1), "v"(aP1) : "memory");
    asm volatile("global_load_async_to_lds_b128 %0, %1, off"
                 :: "v"(lB0), "v"(bP0) : "memory");
    asm volatile("global_load_async_to_lds_b128 %0, %1, off"
                 :: "v"(lB1), "v"(bP1) : "memory");
}

// 16x16x16 MACs = 4 x V_WMMA_F32_16X16X4_F32 from one LDS tile pair
__device__ __forceinline__ v8f tile_mma(const float* sA, const float* sB,
                                        int l15, int half, v8f acc)
{
#pragma unroll
    for (int k4 = 0; k4 < 16; k4 += 4) {
        // A 16x4 frag: lane -> M=l15; VGPR0/1 = K = k4+2*half, +1
        // B 4x16 frag: lane -> N=l15; VGPR0/1 = K rows k4+2*half, +1
        v2f a, b;
        int ka = k4 + 2 * half;
        a[0] = sA[l15 * 16 + ka + 0];
        a[1] = sA[l15 * 16 + ka + 1];
        b[0] = sB[(ka + 0) * 16 + l15];
        b[1] = sB[(ka + 1) * 16 + l15];
        acc = __builtin_amdgcn_wmma_f32_16x16x4_f32(
            false, a, false, b, (short)0, acc, false, false);
    }
    return acc;
}

// ---------------------------------------------------------------------------
// Generic f32 GEMM: C[M,N] = A[M,K] @ B[K,N] (+ bias[N] if bias != null)
// N multiple of 16, K multiple of 32; M arbitrary. OOB A rows are clamped to
// row M-1: they only feed C rows >= M which are never stored, so every load
// is unconditional (EXEC stays all-ones, no divergence, no zero-fill).
// Each wave owns private double-buffered LDS slots -> no inter-wave barriers;
// slot i+1's async batch is in flight while slot i's WMMAs run.
// ---------------------------------------------------------------------------
__global__ __launch_bounds__(256) void gemm_wmma_f32(
    const float* __restrict__ A, const float* __restrict__ B,
    const float* __restrict__ bias, float* __restrict__ C,
    int M, int N, int K)
{
    __shared__ float lA[2][GEMM_WAVES][256];   // [slot][wave][16 rows x 16 k]
    __shared__ float lB[2][GEMM_WAVES][256];   // [slot][wave][16 k x 16 cols]
    const int w    = threadIdx.x >> 5;
    const int lane = threadIdx.x & 31;
    const int half = lane >> 4;             // 0: lanes 0-15, 1: lanes 16-31
    const int l15  = lane & 15;
    const int n0   = blockIdx.x * 16;
    const int m0   = (blockIdx.y * GEMM_WAVES + w) * 16;

    // 256-float tile moved as 2 x (32 lanes x b128): transfer j covers tile
    // element e = j*128 + 4*lane; row = e>>4 (tile-local), col = e&15.
    const int r0 = (4 * lane) >> 4;          // 0..7
    const int c0 = (4 * lane) & 15;          // 0,4,8,12
    const int gmA0 = min(m0 + r0,     M - 1);   // clamped (see note above)
    const int gmA1 = min(m0 + r0 + 8, M - 1);

    const float* aP0 = A + (size_t)gmA0 * K + c0;
    const float* aP1 = A + (size_t)gmA1 * K + c0;
    const float* bP0 = B + (size_t)r0 * N + n0 + c0;
    const float* bP1 = B + (size_t)(r0 + 8) * N + n0 + c0;
    const size_t bStep = (size_t)16 * N;

    const unsigned a0s0 = (unsigned)(uintptr_t)&lA[0][w][4 * lane];
    const unsigned a1s0 = (unsigned)(uintptr_t)&lA[0][w][128 + 4 * lane];
    const unsigned b0s0 = (unsigned)(uintptr_t)&lB[0][w][4 * lane];
    const unsigned b1s0 = (unsigned)(uintptr_t)&lB[0][w][128 + 4 * lane];
    const unsigned a0s1 = (unsigned)(uintptr_t)&lA[1][w][4 * lane];
    const unsigned a1s1 = (unsigned)(uintptr_t)&lA[1][w][128 + 4 * lane];
    const unsigned b0s1 = (unsigned)(uintptr_t)&lB[1][w][4 * lane];
    const unsigned b1s1 = (unsigned)(uintptr_t)&lB[1][w][128 + 4 * lane];

    v8f acc = {};

    // prologue: tile 0 -> slot 0
    async_tile_load(a0s0, a1s0, b0s0, b1s0, aP0, aP1, bP0, bP1);
    aP0 += 16; aP1 += 16; bP0 += bStep; bP1 += bStep;

    for (int kk = 0; kk < K; kk += 32) {
        // issue tile kk+16 -> slot 1 (overlaps with slot-0 compute below)
        asm volatile("s_wait_dscnt 0" ::: "memory");
        async_tile_load(a0s1, a1s1, b0s1, b1s1, aP0, aP1, bP0, bP1);
        aP0 += 16; aP1 += 16; bP0 += bStep; bP1 += bStep;
        asm volatile("s_wait_asynccnt 0x4" ::: "memory");  // drain slot-0 batch only
        acc = tile_mma(&lA[0][w][0], &lB[0][w][0], l15, half, acc);

        if (kk + 32 < K) {
            // issue tile kk+32 -> slot 0 (overlaps with slot-1 compute)
            asm volatile("s_wait_dscnt 0" ::: "memory");
            async_tile_load(a0s0, a1s0, b0s0, b1s0, aP0, aP1, bP0, bP1);
            aP0 += 16; aP1 += 16; bP0 += bStep; bP1 += bStep;
            asm volatile("s_wait_asynccnt 0x4" ::: "memory");
        } else {
            asm volatile("s_wait_asynccnt 0x0" ::: "memory");
        }
        acc = tile_mma(&lA[1][w][0], &lB[1][w][0], l15, half, acc);
    }

    const float bv = (bias != nullptr) ? bias[n0 + l15] : 0.0f;
    float* cP = C + (size_t)(m0 + 8 * half) * N + n0 + l15;
    if (m0 + 16 <= M) {                     // wave-uniform full-tile fast path
#pragma unroll
        for (int r = 0; r < 8; ++r) cP[(size_t)r * N] = acc[r] + bv;
    } else {
#pragma unroll
        for (int r = 0; r < 8; ++r) {
            int row = m0 + r + 8 * half;    // C/D layout: VGPR r -> M=r / r+8
            if (row < M) cP[(size_t)r * N] = acc[r] + bv;
        }
    }
}

// ---------------------------------------------------------------------------
// tiny: et_att[t,h] = sum_d ( (eemb @ We)[t, h*EDIM+d] * ae[h,d] )
// ---------------------------------------------------------------------------
__global__ void etatt_kernel(const float* __restrict__ eemb,
                             const float* __restrict__ We,
                             const float* __restrict__ ae,
                             float* __restrict__ out,
                             int T, int H, int EDIM, int Cdim)
{
    int tid = blockIdx.x * blockDim.x + threadIdx.x;
    if (tid >= T * H) return;
    int t = tid / H, h = tid - t * H;
    float acc = 0.0f;
    for (int d = 0; d < EDIM; ++d) {
        float p = 0.0f;
        for (int c = 0; c < Cdim; ++c)
            p += eemb[t * Cdim + c] * We[(size_t)c * H * EDIM + h * EDIM + d];
        acc += p * ae[h * EDIM + d];
    }
    out[tid] = acc;
}

// el[n,h] = sum_o feat[n,h,o]*al[h,o];  er likewise
__global__ void elr_kernel(const float* __restrict__ feat,
                           const float* __restrict__ al,
                           const float* __restrict__ ar,
                           float* __restrict__ el, float* __restrict__ er,
                           int N, int H, int OUT)
{
    int tid = blockIdx.x * blockDim.x + threadIdx.x;
    if (tid >= N * H) return;
    int n = tid / H, h = tid - n * H;
    const float* f  = feat + ((size_t)n * H + h) * OUT;
    const float* wl = al + h * OUT;
    const float* wr = ar + h * OUT;
    float sl = 0.0f, sr = 0.0f;
    for (int o = 0; o < OUT; ++o) { float v = f[o]; sl += v * wl[o]; sr += v * wr[o]; }
    el[tid] = sl;
    er[tid] = sr;
}

__global__ void fill_kernel(float* __restrict__ p, float v, int n)
{
    int t = blockIdx.x * blockDim.x + threadIdx.x;
    if (t < n) p[t] = v;
}

// sign-bucketed float atomic max (init must be -inf)
__device__ inline void atomicMaxF(float* addr, float val)
{
    val = val + 0.0f;                       // canonicalize -0 -> +0
    if (val >= 0.0f) atomicMax((int*)addr, __float_as_int(val));
    else             atomicMin((unsigned int*)addr, __float_as_uint(val));
}

// pass1: e = leaky(el[src]+er[dst]+et[etype]); store; segment max into mbuf
__global__ void edge_pass1(const int* __restrict__ src, const int* __restrict__ dst,
                           const int* __restrict__ etype,
                           const float* __restrict__ el, const float* __restrict__ er,
                           const float* __restrict__ etatt,
                           float* __restrict__ e_out, float* __restrict__ mbuf,
                           int E, int H)
{
    int tid = blockIdx.x * blockDim.x + threadIdx.x;
    if (tid >= E * H) return;
    int i = tid / H, h = tid - i * H;
    int s = src[i], d = dst[i], t = etype[i];
    float e = el[s * H + h] + er[d * H + h] + etatt[t * H + h];
    e = (e >= 0.0f) ? e : SLOPE_C * e;
    e_out[tid] = e;
    atomicMaxF(&mbuf[d * H + h], e);
}

// pass2: ex = exp(e - m[dst]); store; segment sum into sbuf
__global__ void edge_pass2(const int* __restrict__ dst,
                           float* __restrict__ e, const float* __restrict__ mbuf,
                           float* __restrict__ sbuf, int E, int H)
{
    int tid = blockIdx.x * blockDim.x + threadIdx.x;
    if (tid >= E * H) return;
    int i = tid / H, h = tid - i * H;
    int d = dst[i];
    float ex = expf(e[tid] - mbuf[d * H + h]);
    e[tid] = ex;
    atomicAdd(&sbuf[d * H + h], ex);
}

// pass3: a = ex / s[dst]; optional residual-attention blend; optional save
__global__ void edge_pass3(const int* __restrict__ dst,
                           float* __restrict__ e, const float* __restrict__ sbuf,
                           const float* __restrict__ resattn, float* __restrict__ asave,
                           int E, int H)
{
    int tid = blockIdx.x * blockDim.x + threadIdx.x;
    if (tid >= E * H) return;
    int i = tid / H, h = tid - i * H;
    int d = dst[i];
    float a = e[tid] / sbuf[d * H + h];
    if (resattn) a = a * (1.0f - ALPHA_C) + resattn[tid] * ALPHA_C;
    e[tid] = a;
    if (asave) asave[tid] = a;
}

// rst[dst,h,o] += feat[src,h,o] * a[i,h]   (thread per (edge, h, o))
// A 256-thread block covers one full edge for H*OUT==256: src/dst broadcast,
// feat reads and atomic scatters are contiguous 1KB runs.
__global__ void edge_aggregate(const int* __restrict__ src, const int* __restrict__ dst,
                               const float* __restrict__ a, const float* __restrict__ feat,
                               float* __restrict__ rst, int total, int H, int OUT)
{
    int tid = blockIdx.x * blockDim.x + threadIdx.x;
    if (tid >= total) return;
    int HO = H * OUT;
    int i = tid / HO;
    int r = tid - i * HO;
    int h = r / OUT;
    int s = src[i], d = dst[i];
    atomicAdd(&rst[(size_t)d * HO + r], feat[(size_t)s * HO + r] * a[i * H + h]);
}

__global__ void elu_kernel(const float* __restrict__ x, float* __restrict__ y, int n)
{
    int t = blockIdx.x * blockDim.x + threadIdx.x;
    if (t < n) { float v = x[t]; y[t] = (v > 0.0f) ? v : expm1f(v); }
}

// ---------------------------------------------------------------------------
// host-side orchestration
// ---------------------------------------------------------------------------
static inline void launch_gemm(const float* A, const float* B, const float* bias,
                               float* C, int M, int N, int K, hipStream_t s)
{
    dim3 grid(N / 16, ((M + 15) / 16 + GEMM_WAVES - 1) / GEMM_WAVES);
    gemm_wmma_f32<<<grid, 256, 0, s>>>(A, B, bias, C, M, N, K);
}

static inline void launch_fill(float* p, float v, int n, hipStream_t s)
{
    fill_kernel<<<(n + 255) / 256, 256, 0, s>>>(p, v, n);
}

extern "C" void kernel_launch(void* const* d_in, const int* in_sizes, int n_in,
                              void* d_out, int out_size, void* d_ws, size_t ws_size,
                              hipStream_t stream)
{
    (void)in_sizes; (void)n_in; (void)out_size; (void)ws_size;

    const float* x0    = (const float*)d_in[0];
    const float* x1    = (const float*)d_in[1];
    const float* x2    = (const float*)d_in[2];
    const float* fcw0  = (const float*)d_in[3];
    const float* fcb0  = (const float*)d_in[4];
    const float* fcw1  = (const float*)d_in[5];
    const float* fcb1  = (const float*)d_in[6];
    const float* fcw2  = (const float*)d_in[7];
    const float* fcb2  = (const float*)d_in[8];
    const float* W0    = (const float*)d_in[9];
    const float* We0   = (const float*)d_in[10];
    const float* eemb0 = (const float*)d_in[11];
    const float* al0   = (const float*)d_in[12];
    const float* ar0   = (const float*)d_in[13];
    const float* ae0   = (const float*)d_in[14];
    const float* W1    = (const float*)d_in[15];
    const float* We1   = (const float*)d_in[16];
    const float* eemb1 = (const float*)d_in[17];
    const float* al1   = (const float*)d_in[18];
    const float* ar1   = (const float*)d_in[19];
    const float* ae1   = (const float*)d_in[20];
    const float* resW1 = (const float*)d_in[21];
    const float* W2    = (const float*)d_in[22];
    const float* We2   = (const float*)d_in[23];
    const float* eemb2 = (const float*)d_in[24];
    const float* al2   = (const float*)d_in[25];
    const float* ar2   = (const float*)d_in[26];
    const float* ae2   = (const float*)d_in[27];
    const float* resW2 = (const float*)d_in[28];
    const int*   src   = (const int*)d_in[29];
    const int*   dst   = (const int*)d_in[30];
    const int*   etype = (const int*)d_in[31];

    const int N = 50000, E = 800000;

    // workspace layout (floats)
    float* W = (float*)d_ws;
    size_t off = 0;
    auto alloc = [&](size_t n) { float* p = W + off; off += (n + 63) & ~(size_t)63; return p; };
    float* h0    = alloc((size_t)N * 64);     // layer-0 input features
    float* hbig  = alloc((size_t)N * 256);    // h1 then h2
    float* feat  = alloc((size_t)N * 256);    // per-layer feat = h @ W
    float* rst   = alloc((size_t)N * 256);    // aggregation accumulator
    float* el    = alloc((size_t)N * 8);
    float* er    = alloc((size_t)N * 8);
    float* mbuf  = alloc((size_t)N * 8);
    float* sbuf  = alloc((size_t)N * 8);
    float* et0   = alloc(64);
    float* et1   = alloc(64);
    float* et2   = alloc(64);
    float* eedge = alloc((size_t)E * 8);      // e -> ex -> a
    float* aprev = alloc((size_t)E * 8);      // layer-0 attention (residual)
    float* out   = (float*)d_out;             // [50000, 16]

    // edge-type attention terms (tiny)
    etatt_kernel<<<1, 64, 0, stream>>>(eemb0, We0, ae0, et0, 6, 8, 64, 64);
    etatt_kernel<<<1, 64, 0, stream>>>(eemb1, We1, ae1, et1, 6, 8, 64, 64);
    etatt_kernel<<<1, 64, 0, stream>>>(eemb2, We2, ae2, et2, 6, 1, 64, 64);

    // per-type input projections -> h0 [50000, 64]
    launch_gemm(x0, fcw0, fcb0, h0,              20000, 64, 128, stream);
    launch_gemm(x1, fcw1, fcb1, h0 + 20000 * 64, 17000, 64, 256, stream);
    launch_gemm(x2, fcw2, fcb2, h0 + 37000 * 64, 13000, 64,  64, stream);

    const int NH  = N * 8;
    const int EH  = E * 8;
    const int NHO = N * 256;
    const int EHO = E * 256;

    // ---------------- layer 0: 64 -> (8,32), no residual, activate ----------
    launch_gemm(h0, W0, nullptr, feat, N, 256, 64, stream);
    elr_kernel<<<(NH + 255) / 256, 256, 0, stream>>>(feat, al0, ar0, el, er, N, 8, 32);
    launch_fill(rst, 0.0f, NHO, stream);
    launch_fill(mbuf, -INFINITY, NH, stream);
    launch_fill(sbuf, 0.0f, NH, stream);
    edge_pass1<<<(EH + 255) / 256, 256, 0, stream>>>(src, dst, etype, el, er, et0, eedge, mbuf, E, 8);
    edge_pass2<<<(EH + 255) / 256, 256, 0, stream>>>(dst, eedge, mbuf, sbuf, E, 8);
    edge_pass3<<<(EH + 255) / 256, 256, 0, stream>>>(dst, eedge, sbuf, nullptr, aprev, E, 8);
    edge_aggregate<<<(EHO + 255) / 256, 256, 0, stream>>>(src, dst, eedge, feat, rst, EHO, 8, 32);
    elu_kernel<<<(NHO + 255) / 256, 256, 0, stream>>>(rst, hbig, NHO);   // h1

    // ---------------- layer 1: 256 -> (8,32), residual + res_attn, activate -
    launch_gemm(hbig, W1, nullptr, feat, N, 256, 256, stream);
    elr_kernel<<<(NH + 255) / 256, 256, 0, stream>>>(feat, al1, ar1, el, er, N, 8, 32);
    launch_gemm(hbig, resW1, nullptr, rst, N, 256, 256, stream);         // rst = h1 @ resW1
    launch_fill(mbuf, -INFINITY, NH, stream);
    launch_fill(sbuf, 0.0f, NH, stream);
    edge_pass1<<<(EH + 255) / 256, 256, 0, stream>>>(src, dst, etype, el, er, et1, eedge, mbuf, E, 8);
    edge_pass2<<<(EH + 255) / 256, 256, 0, stream>>>(dst, eedge, mbuf, sbuf, E, 8);
    edge_pass3<<<(EH + 255) / 256, 256, 0, stream>>>(dst, eedge, sbuf, aprev, nullptr, E, 8);
    edge_aggregate<<<(EHO + 255) / 256, 256, 0, stream>>>(src, dst, eedge, feat, rst, EHO, 8, 32);
    elu_kernel<<<(NHO + 255) / 256, 256, 0, stream>>>(rst, hbig, NHO);   // h2 (overwrite)

    // ---------------- layer 2: 256 -> (1,16), residual, no activation -------
    launch_gemm(hbig, W2, nullptr, feat, N, 16, 256, stream);
    elr_kernel<<<(N + 255) / 256, 256, 0, stream>>>(feat, al2, ar2, el, er, N, 1, 16);
    launch_gemm(hbig, resW2, nullptr, out, N, 16, 256, stream);          // out = h2 @ resW2
    launch_fill(mbuf, -INFINITY, N, stream);
    launch_fill(sbuf, 0.0f, N, stream);
    edge_pass1<<<(E + 255) / 256, 256, 0, stream>>>(src, dst, etype, el, er, et2, eedge, mbuf, E, 1);
    edge_pass2<<<(E + 255) / 256, 256, 0, stream>>>(dst, eedge, mbuf, sbuf, E, 1);
    edge_pass3<<<(E + 255) / 256, 256, 0, stream>>>(dst, eedge, sbuf, nullptr, nullptr, E, 1);
    edge_aggregate<<<(E * 16 + 255) / 256, 256, 0, stream>>>(src, dst, eedge, feat, out, E * 16, 1, 16);
}